// RobustLoss_42030549958903
// MI455X (gfx1250) — compile-verified
//
#include <hip/hip_runtime.h>
#include <math.h>
#include <stdint.h>

// ---------------------------------------------------------------------------
// RobustLoss (CVaR, size=0.1, reg=0.1, tol=0.01) for MI455X / gfx1250.
//
// Phase 1 (bandwidth-bound, 524 MB @ 23.3 TB/s ~= 22.5 us floor):
//   per-row cross-entropy with async global->LDS staging (CDNA5 async path),
//   non-temporal (th:TH_LOAD_NT) since logits are read once and exceed the
//   192 MB L2 -- keeps v/w scratch resident for phases 2-3.
//   Exact two-pass softmax over the LDS-resident row (1 exp per element).
// Phase 2: w[i] = exp((v[i]-vmax)/REG) so every bisection evaluation is a
//   scalar-broadcast mul+min over L2/LDS-resident data (no transcendentals
//   in the loop).
// Phase 3: faithful expansion+bisection + final loss in one workgroup, with
//   156 KB of w cached in LDS (async-filled) to cut per-evaluation L2 traffic.
// ---------------------------------------------------------------------------

#define WAVES_PER_BLOCK 8
#define ROWS_PER_WAVE   4
#define MAX_CHUNKS      8                      // supports C <= 1024 (C/4 <= 256)
#define ROW_F4          (32 * MAX_CHUNKS)      // 256 float4 slots = 4096 B
#define INV_REG         10.0f                  // 1/REG
#define REGL            0.1f
#define CAP             10.0f                  // 1/alpha
#define TOLF            0.01f
#define K3_LDS_N        39936                  // floats of w cached in LDS (156 KB)

__device__ __forceinline__ float wave_max(float x) {
#pragma unroll
  for (int o = 16; o > 0; o >>= 1) x = fmaxf(x, __shfl_xor(x, o, 32));
  return x;
}
__device__ __forceinline__ float wave_sum(float x) {
#pragma unroll
  for (int o = 16; o > 0; o >>= 1) x += __shfl_xor(x, o, 32);
  return x;
}

__global__ void init_kernel(float* ws) {
  ws[0] = __int_as_float(0);  // vmax bits (CE > 0, so int-max on bits is valid)
  ws[1] = 0.0f;               // S0 accumulator
}

// ---------------- Phase 1: per-row CE via async LDS staging ----------------
__global__ __launch_bounds__(256) void ce_rows_kernel(
    const float* __restrict__ outputs, const int* __restrict__ targets,
    float* __restrict__ v, int* __restrict__ vmax_bits, int N, int C) {
  __shared__ __align__(16) float lds[WAVES_PER_BLOCK * 2 * ROW_F4 * 4];  // 64 KB
  const int lane    = threadIdx.x & 31;
  const int wave    = threadIdx.x >> 5;
  const int nchunks = C >> 2;  // float4s per row (C multiple of 4, <=1024)
  float* wbuf = &lds[wave * 2 * ROW_F4 * 4];

  const long long rowBase =
      (long long)(blockIdx.x * WAVES_PER_BLOCK + wave) * ROWS_PER_WAVE;
  if (rowBase >= (long long)N) return;  // no barriers in this kernel: safe
  int nrows = ROWS_PER_WAVE;
  if (rowBase + nrows > (long long)N) nrows = (int)((long long)N - rowBase);

  // Issue exactly MAX_CHUNKS wave-wide async instructions per row (tail lanes
  // clamp to a valid in-row chunk) so ASYNCcnt bookkeeping is compile-time.
  auto issue = [&](long long row, int buf) {
    const float* src = outputs + row * (long long)C;
    unsigned ldsbase = (unsigned)(uintptr_t)(&wbuf[buf * ROW_F4 * 4]);
#pragma unroll
    for (int j = 0; j < MAX_CHUNKS; ++j) {
      int idx = lane + 32 * j;
      int idxc = idx < nchunks ? idx : (nchunks - 1);
      unsigned laddr = ldsbase + (unsigned)idxc * 16u;
      unsigned long long gaddr = (unsigned long long)(uintptr_t)(src + idxc * 4);
      // CDNA5 async copy: 16B/lane, wave-wide, ASYNCcnt-tracked, non-temporal
      // (read-once 524 MB stream; don't let it sweep the 192 MB L2).
      asm volatile("global_load_async_to_lds_b128 %0, %1, off th:TH_LOAD_NT"
                   :: "v"(laddr), "v"(gaddr) : "memory");
    }
  };

  float vloc = 0.0f;  // per-wave running max of v (lane 0 only is meaningful)

  auto process = [&](long long row, int buf) {
    const float4* b = (const float4*)&wbuf[buf * ROW_F4 * 4];
    float m = -3.402823466e+38f;
#pragma unroll
    for (int j = 0; j < MAX_CHUNKS; ++j) {
      int idx = lane + 32 * j;
      if (idx < nchunks) {
        float4 x = b[idx];
        m = fmaxf(m, fmaxf(fmaxf(x.x, x.y), fmaxf(x.z, x.w)));
      }
    }
    m = wave_max(m);
    float s = 0.0f;
#pragma unroll
    for (int j = 0; j < MAX_CHUNKS; ++j) {
      int idx = lane + 32 * j;
      if (idx < nchunks) {
        float4 x = b[idx];
        s += __expf(x.x - m) + __expf(x.y - m) + __expf(x.z - m) + __expf(x.w - m);
      }
    }
    s = wave_sum(s);
    int t = targets[row];
    float xt = ((const float*)b)[t];  // uniform address -> LDS broadcast read
    if (lane == 0) {
      float vv = m + __logf(s) - xt;  // CE = max + log(sum exp(x-max)) - x[t]
      v[row] = vv;
      vloc = fmaxf(vloc, vv);
    }
  };

  issue(rowBase, 0);
  for (int k = 0; k < nrows; ++k) {
    if (k + 1 < nrows) {
      issue(rowBase + k + 1, (k + 1) & 1);
      asm volatile("s_wait_asynccnt 8" ::: "memory");  // batch k done; k+1 in flight
    } else {
      asm volatile("s_wait_asynccnt 0" ::: "memory");
    }
    process(rowBase + k, k & 1);
  }
  if (lane == 0) atomicMax(vmax_bits, __float_as_int(vloc));  // 1 atomic / wave
}

// --------- Phase 2: w[i] = exp((v-vmax)/REG), S0 = sum(w) ----------
__global__ __launch_bounds__(256) void weights_kernel(
    const float* __restrict__ v, float* __restrict__ w,
    const int* __restrict__ vmax_bits, float* __restrict__ S0, int N) {
  __shared__ float red[8];
  const float vmax = __int_as_float(*vmax_bits);
  float acc = 0.0f;
  for (int i = blockIdx.x * blockDim.x + threadIdx.x; i < N;
       i += gridDim.x * blockDim.x) {
    float wi = __expf((v[i] - vmax) * INV_REG);
    w[i] = wi;
    acc += wi;
  }
  acc = wave_sum(acc);
  int lane = threadIdx.x & 31, wid = threadIdx.x >> 5;
  if (lane == 0) red[wid] = acc;
  __syncthreads();
  if (wid == 0) {
    float a = (lane < (int)(blockDim.x >> 5)) ? red[lane] : 0.0f;
    a = wave_sum(a);
    if (lane == 0) atomicAdd(S0, a);
  }
}

// --------- Phase 3: expansion + bisection + final loss (one WG) ----------
__global__ __launch_bounds__(1024) void bisect_kernel(
    const float* __restrict__ v, const float* __restrict__ w,
    const int* __restrict__ vmax_bits, const float* __restrict__ S0p,
    float* __restrict__ out, int N) {
  __shared__ __align__(16) float w_lds[K3_LDS_N];  // 156 KB w cache
  __shared__ float red[32];
  __shared__ float bc;
  const int tid = threadIdx.x, lane = tid & 31, wid = tid >> 5;
  const float vmax = __int_as_float(*vmax_bits);
  const float S0 = *S0p;
  const float m = (float)N;
  const int ldsn = (N < K3_LDS_N) ? (N & ~3) : K3_LDS_N;

  // Fill the LDS w-cache with async b128 copies, then publish to all waves.
  {
    unsigned ldsbase = (unsigned)(uintptr_t)(&w_lds[0]);
    const int nchunk = ldsn >> 2;
    for (int c = tid; c < nchunk; c += 1024) {
      unsigned laddr = ldsbase + (unsigned)c * 16u;
      unsigned long long gaddr = (unsigned long long)(uintptr_t)(w + c * 4);
      asm volatile("global_load_async_to_lds_b128 %0, %1, off"
                   :: "v"(laddr), "v"(gaddr) : "memory");
    }
    asm volatile("s_wait_asynccnt 0" ::: "memory");
    __syncthreads();
  }

  auto block_sum = [&](float a) -> float {
    a = wave_sum(a);
    if (lane == 0) red[wid] = a;
    __syncthreads();
    if (tid == 0) {
      float t = 0.0f;
      for (int i = 0; i < 32; ++i) t += red[i];
      bc = t;
    }
    __syncthreads();
    float r = bc;
    __syncthreads();
    return r;
  };

  // f(eta) = 1 - sum(min(w*T, CAP))/m,  T = exp((vmax-eta)/REG).
  // T may overflow to +inf at very low eta: w*inf = inf (or NaN when w==0);
  // fminf(NaN,CAP)=CAP which matches the capped branch there.
  auto f_of = [&](float eta) -> float {
    float T = __expf((vmax - eta) * INV_REG);
    float a = 0.0f;
    for (int i = tid; i < ldsn; i += 1024) a += fminf(w_lds[i] * T, CAP);
    for (int i = ldsn + tid; i < N; i += 1024) a += fminf(w[i] * T, CAP);
    return 1.0f - block_sum(a) / m;
  };

  // eta_min = REG*(logsumexp(v/REG) - log m) = vmax + REG*(log S0 - log m)
  float lo = vmax + REGL * (__logf(S0) - __logf(m));
  float hi = vmax;

  // dynamic interval expansion (reference: while f(lo)>0 || f(hi)<0)
  for (int it = 0; it < 64; ++it) {
    float fl = f_of(lo), fh = f_of(hi);
    if (!((fl > 0.0f) || (fh < 0.0f))) break;
    float iv = hi - lo;
    if (fl > 0.0f) { hi = lo; lo = lo - 2.0f * iv; }
    else           { lo = hi; hi = hi + 2.0f * iv; }
  }
  // bisection until |f(mid)| <= tol or 500 iters (exit before updating bounds)
  for (int it = 0; it < 500; ++it) {
    float eta = 0.5f * (lo + hi);
    float val = f_of(eta);
    if (fabsf(val) <= TOLF) break;
    if (val > 0.0f) hi = eta; else lo = eta;
  }

  float Ts = __expf((vmax - 0.5f * (lo + hi)) * INV_REG);
  float ad = 0.0f, ak = 0.0f;
  for (int i = tid; i < N; i += 1024) {
    float wi = (i < ldsn) ? w_lds[i] : w[i];
    float p = fminf(wi * Ts, CAP) / m;
    ad += p * v[i];
    ak += p * __logf(p + 1e-10f);
  }
  float dot = block_sum(ad);
  float kl  = block_sum(ak);
  if (tid == 0) out[0] = dot - REGL * (__logf(m) + kl);
}

extern "C" void kernel_launch(void* const* d_in, const int* in_sizes, int n_in,
                              void* d_out, int out_size, void* d_ws, size_t ws_size,
                              hipStream_t stream) {
  const float* outputs = (const float*)d_in[0];
  const int*   targets = (const int*)d_in[1];
  const int N = in_sizes[1];
  const int C = in_sizes[0] / N;

  float* ws = (float*)d_ws;
  int*   vmax_bits = (int*)ws;   // ws[0]
  float* S0 = ws + 1;            // ws[1]
  float* v  = ws + 16;           // [N]
  float* w  = v + N;             // [N]

  init_kernel<<<1, 1, 0, stream>>>(ws);

  const int rowsPerBlock = WAVES_PER_BLOCK * ROWS_PER_WAVE;  // 32
  const int blocks = (N + rowsPerBlock - 1) / rowsPerBlock;
  ce_rows_kernel<<<blocks, 256, 0, stream>>>(outputs, targets, v, vmax_bits, N, C);

  weights_kernel<<<1024, 256, 0, stream>>>(v, w, vmax_bits, S0, N);

  bisect_kernel<<<1, 1024, 0, stream>>>(v, w, vmax_bits, S0, (float*)d_out, N);
}